// VoxelSampler_6657199309190
// MI455X (gfx1250) — compile-verified
//
#include <hip/hip_runtime.h>
#include <math.h>

// ---------------------------------------------------------------------------
// VoxelSampler (cylindrical pool) for MI455X / gfx1250, wave32.
//
//   d2(box m, point n) = A_m . B_n  with
//     A_m = [-2*cx, -2*cy, cx^2+cy^2, 1]        (16x4 f32 A-matrix)
//     B_n = [px, py, 1, px^2+py^2]              (4x16 f32 B-matrix)
//   -> V_WMMA_F32_16X16X4_F32 computes a 16x16 tile of squared distances.
// ---------------------------------------------------------------------------

typedef __attribute__((ext_vector_type(2))) float v2f;
typedef __attribute__((ext_vector_type(8))) float v8f;

#define VS_GAMMA 1.1f

// ---------------------------------------------------------------------------
// Phase 1: per-(box,point) in-radius mask via WMMA, packed to a bitmap.
// mask16[b * nt16e + t] bit i  <=>  point (t*16 + i) inside cylinder of box b.
// grid = (ceil(B/16), PTILE_BLOCKS), block = 256 (8 waves).
// ---------------------------------------------------------------------------
__global__ __launch_bounds__(256) void vs_phase1_mask(
    const float* __restrict__ pts,     // [N,5]
    const float* __restrict__ boxes,   // [B,7] cx,cy,cz,dx,dy,dz,heading
    unsigned short* __restrict__ mask16,
    int N, int B, int nt16e)           // nt16e = ceil(N/16) rounded up to even
{
  const int lane = threadIdx.x & 31;
  const int wave = threadIdx.x >> 5;
  const int half = lane >> 4;          // 0: lanes 0-15, 1: lanes 16-31
  const int l15  = lane & 15;
  const int boxTile = blockIdx.x * 16;

  // ----- A matrix (16x4 f32): VGPR0 = K=0 (lo lanes) / K=2 (hi lanes),
  //                            VGPR1 = K=1 (lo lanes) / K=3 (hi lanes).
  int m = boxTile + l15; if (m >= B) m = B - 1;
  float cx = boxes[m * 7 + 0];
  float cy = boxes[m * 7 + 1];
  v2f A;
  A.x = half ? fmaf(cx, cx, cy * cy) : (-2.0f * cx);   // K=2 : K=0
  A.y = half ? 1.0f                  : (-2.0f * cy);   // K=3 : K=1

  // Per-D-row radius^2: D VGPR rr, lane-half h holds box (boxTile + rr + 8h).
  float r2row[8];
#pragma unroll
  for (int rr = 0; rr < 8; ++rr) {
    int bm = boxTile + rr + half * 8; if (bm >= B) bm = B - 1;
    float hx = boxes[bm * 7 + 3] * 0.5f;
    float hy = boxes[bm * 7 + 4] * 0.5f;
    float rad = sqrtf(fmaf(hx, hx, hy * hy)) * VS_GAMMA;
    r2row[rr] = rad * rad;
  }

  // Per-lane mask-store row: lane l (<16) owns box boxTile+l.
  const bool storeLane = (l15 == lane) && (boxTile + lane < B); // lanes 0-15, in range
  unsigned short* rowPtr =
      mask16 + (size_t)((boxTile + l15 < B) ? (boxTile + l15) : 0) * nt16e;

  const int wavesPerBoxTile = gridDim.y * (blockDim.x >> 5);
  const int waveGlobal      = blockIdx.y * (blockDim.x >> 5) + wave;

  for (int t = waveGlobal; t < nt16e; t += wavesPerBoxTile) {   // wave-uniform
    int  n  = t * 16 + l15;
    int  nc = (n < N) ? n : (N - 1);         // clamp -> unconditional load
    float px0 = pts[(size_t)nc * 5 + 0];
    float py0 = pts[(size_t)nc * 5 + 1];
    // Out-of-range points pushed to a huge finite distance -> mask bit 0.
    float px = (n < N) ? px0 : 2.0e18f;
    float py = (n < N) ? py0 : 2.0e18f;

    v2f Bm;                                  // 4x16 B matrix, mirrored layout
    Bm.x = half ? 1.0f                  : px;   // K=2 : K=0
    Bm.y = half ? fmaf(px, px, py * py) : py;   // K=3 : K=1

    v8f C = {};
    // D = A x B (+0): 16x16 tile of squared distances.
    C = __builtin_amdgcn_wmma_f32_16x16x4_f32(
        /*neg_a=*/false, A, /*neg_b=*/false, Bm,
        /*c_mod=*/(short)0, C, /*reuse_a=*/false, /*reuse_b=*/false);

    // 8 ballots: bal[rr] bits 0-15 -> box boxTile+rr, bits 16-31 -> +rr+8,
    // bit i = point t*16+i in-radius.
    unsigned bal[8];
#pragma unroll
    for (int rr = 0; rr < 8; ++rr)
      bal[rr] = __builtin_amdgcn_ballot_w32(C[rr] <= r2row[rr]);

    // Route ballot (lane&7) to each lane via a 3-level select tree, pick the
    // half for boxes 0-7 vs 8-15, then one exec-masked b16 store covers the
    // whole 16-box tile (no scalar branching).
    const int idx = lane & 7;
    unsigned c01 = (idx & 1) ? bal[1] : bal[0];
    unsigned c23 = (idx & 1) ? bal[3] : bal[2];
    unsigned c45 = (idx & 1) ? bal[5] : bal[4];
    unsigned c67 = (idx & 1) ? bal[7] : bal[6];
    unsigned c03 = (idx & 2) ? c23 : c01;
    unsigned c47 = (idx & 2) ? c67 : c45;
    unsigned cc  = (idx & 4) ? c47 : c03;
    unsigned short hw = (unsigned short)((lane & 8) ? (cc >> 16) : (cc & 0xFFFFu));
    if (storeLane) rowPtr[t] = hw;
  }
}

// ---------------------------------------------------------------------------
// Phase 2: per-box ordered compaction. One wave32 per box scans the bitmap in
// index order, wave-scan of popcounts assigns ranks, first S hits are gathered
// (lowest index first == torch/jax top_k tie-breaking), rest stays zero.
// grid = B, block = 32.
// ---------------------------------------------------------------------------
__global__ __launch_bounds__(32) void vs_phase2_select(
    const float* __restrict__ pts,            // [N,5]
    const unsigned int* __restrict__ maskw,   // [B][nw] 32 points per word
    float* __restrict__ out,                  // [B,S,5]
    int S, int nw)
{
  const int b    = blockIdx.x;
  const int lane = threadIdx.x;

  float* ob = out + (size_t)b * S * 5;
  for (int i = lane; i < S * 5; i += 32) ob[i] = 0.0f;   // zero-fill slots

  const unsigned int* mw = maskw + (size_t)b * nw;
  int total = 0;
  for (int base = 0; base < nw; base += 32) {
    int wi  = base + lane;
    int wic = (wi < nw) ? wi : (nw - 1);      // clamp -> unconditional load
    unsigned int w = mw[wic];
    w = (wi < nw) ? w : 0u;
    int c = __popc(w);

    // Inclusive wave scan of popcounts.
    int inc = c;
#pragma unroll
    for (int off = 1; off < 32; off <<= 1) {
      int t = __shfl_up(inc, off, 32);
      if (lane >= off) inc += t;
    }
    int excl      = inc - c;
    int waveTotal = __shfl(inc, 31, 32);   // uniform across the wave

    // Emit this lane's set bits at their global ranks.
    int rank = total + excl;
    unsigned int ww = w;
    while (ww && rank < S) {
      int bit = __ffs(ww) - 1;
      ww &= ww - 1u;
      int pidx = wi * 32 + bit;
#pragma unroll
      for (int f = 0; f < 5; ++f)
        ob[rank * 5 + f] = pts[(size_t)pidx * 5 + f];
      ++rank;
    }

    total += waveTotal;                    // uniform
    if (total >= S) break;                 // uniform early exit
  }
}

// ---------------------------------------------------------------------------
extern "C" void kernel_launch(void* const* d_in, const int* in_sizes, int n_in,
                              void* d_out, int out_size, void* d_ws, size_t ws_size,
                              hipStream_t stream) {
  const float* pts   = (const float*)d_in[0];   // [N,5] f32
  const float* boxes = (const float*)d_in[1];   // [B,7] f32
  (void)n_in; (void)ws_size;

  const int N = in_sizes[0] / 5;
  const int B = in_sizes[1] / 7;
  const int S = out_size / (B * 5);             // num_sample (128)

  const int nt16  = (N + 15) / 16;
  const int nt16e = (nt16 + 1) & ~1;            // even # of 16-point tiles
  unsigned short* mask16 = (unsigned short*)d_ws;   // B * nt16e * 2 bytes

  // Phase 1: in-radius bitmap via f32 WMMA (16 boxes x 16 points per wmma).
  dim3 g1((B + 15) / 16, 52);                   // 16 box tiles x 52 point blocks
  vs_phase1_mask<<<g1, 256, 0, stream>>>(pts, boxes, mask16, N, B, nt16e);

  // Phase 2: ordered capped compaction, one wave32 per box.
  vs_phase2_select<<<B, 32, 0, stream>>>(pts, (const unsigned int*)d_ws,
                                         (float*)d_out, S, nt16e / 2);
}